// DifferentiableKF_86449101734297
// MI455X (gfx1250) — compile-verified
//
#include <hip/hip_runtime.h>

// ---------------------------------------------------------------------------
// Differentiable KF innovation-covariance scan for MI455X (gfx1250).
//
// Chunked prefix-scan across NB~250 WGPs (the recurrence is linear/affine):
//   B1) M, CM=C@M, CM2=C@M^2, CK=C@K, MK=M@K, CMK=C@M@K, M2, M^L  (1 block)
//   A)  per-chunk Horner reduction of the x-recurrence            (parallel)
//   B2) chunk-start states x_start / P_start; P chain uses WMMA   (1 block)
//   C)  heavy pass, 2 scan steps per barrier:
//         cp1 = CM@P, cp2 = CM2@P, P <- M2@P share ONE set of B fragments
//       (40 fp32 WMMAs / 2 steps, 1 barrier / 2 steps). x/eps are computed
//       as composite maps of BATCH-START state only:
//         e1 = oc1 - CM x - CK op1
//         e2 = oc2 - CM2 x - CMK op1 - CK oc1
//         x' = M2 x + MK op1 + K oc1
//       -> one readlane broadcast round per batch, 6 independent FMA chains
//       (issue-bound, not latency-bound), zero LDS traffic.
//   D)  fixed-order reduction + initial term + Cholesky + 2*sum(log diag).
// Fallback: single-block serial kernel if workspace is insufficient.
// ---------------------------------------------------------------------------

#define NS   32
#define LDP  33      // padded LDS stride; gcd(33,64)=1 -> transposed reads conflict-free
#define TILE 16

typedef __attribute__((ext_vector_type(2))) float v2f;
typedef __attribute__((ext_vector_type(8))) float v8f;

__device__ __forceinline__ v8f wmma4(v2f a, v2f b, v8f c) {
  // D = A(16x4) * B(4x16) + C(16x16), fp32 WMMA
  return __builtin_amdgcn_wmma_f32_16x16x4_f32(false, a, false, b, (short)0, c,
                                               false, false);
}

// uniform-lane broadcast: v_readlane_b32 (no LDS traffic, unlike __shfl)
__device__ __forceinline__ float bcastf(float v, int l) {
  return __uint_as_float(__builtin_amdgcn_readlane(__float_as_uint(v), l));
}

// B fragment (4x16) for K-chunk kc from a row-major [NS x LDP] LDS buffer.
__device__ __forceinline__ v2f loadB(const float* buf, int kc, int tn, int l16,
                                     int lhalf) {
  const int kk = kc * 4 + 2 * lhalf;
  v2f b;
  b.x = buf[kk * LDP + tn * TILE + l16];
  b.y = buf[(kk + 1) * LDP + tn * TILE + l16];
  return b;
}

// Scatter a 16x16 C/D accumulator tile to a row-major [NS x LDP] buffer.
__device__ __forceinline__ void storeD(float* buf, v8f d, int tm, int tn,
                                       int l16, int lhalf) {
#pragma unroll
  for (int j = 0; j < 8; ++j)
    buf[(tm * TILE + j + 8 * lhalf) * LDP + tn * TILE + l16] = d[j];
}

// d = a @ b  (all [NS x LDP] LDS, 128 threads, caller provides barriers)
__device__ __forceinline__ void mm_valu(float* d, const float* a,
                                        const float* b, int tid) {
  for (int e = tid; e < NS * NS; e += 128) {
    const int i = e >> 5, j = e & 31;
    float s = 0.f;
#pragma unroll 8
    for (int k = 0; k < NS; ++k) s += a[i * LDP + k] * b[k * LDP + j];
    d[i * LDP + j] = s;
  }
}

// ============================ Kernel B1 ====================================
// Constant matrices -> workspace: M, CM, CM2, CK, MK, CMK, M2, ML=M^L
__global__ __launch_bounds__(128, 1) void kf_prep_kernel(
    const float* __restrict__ A, const float* __restrict__ C,
    const float* __restrict__ K, float* __restrict__ Mg,
    float* __restrict__ CMg, float* __restrict__ M2g, float* __restrict__ CM2g,
    float* __restrict__ CKg, float* __restrict__ MKg, float* __restrict__ CMKg,
    float* __restrict__ MLg, int L) {
  __shared__ float Ml[NS * LDP], Cl[NS * LDP], Kl[NS * LDP];
  __shared__ float T0[NS * LDP], T1[NS * LDP], R[NS * LDP], B[NS * LDP];
  const int tid = threadIdx.x;

  for (int e = tid; e < NS * NS; e += 128) {
    const int i = e >> 5, j = e & 31;
    Cl[i * LDP + j] = C[e];
    Kl[i * LDP + j] = K[e];
  }
  __syncthreads();
  for (int e = tid; e < NS * NS; e += 128) {
    const int i = e >> 5, j = e & 31;
    float s = A[e];
#pragma unroll 8
    for (int k = 0; k < NS; ++k) s -= Kl[i * LDP + k] * Cl[k * LDP + j];
    Ml[i * LDP + j] = s;
    Mg[e] = s;
  }
  __syncthreads();

  mm_valu(T0, Cl, Ml, tid);  // T0 = CM (kept live)
  __syncthreads();
  for (int e = tid; e < NS * NS; e += 128)
    CMg[e] = T0[(e >> 5) * LDP + (e & 31)];

  mm_valu(T1, T0, Ml, tid);  // CM2 = CM@M
  __syncthreads();
  for (int e = tid; e < NS * NS; e += 128)
    CM2g[e] = T1[(e >> 5) * LDP + (e & 31)];
  __syncthreads();

  mm_valu(T1, Cl, Kl, tid);  // CK = C@K
  __syncthreads();
  for (int e = tid; e < NS * NS; e += 128)
    CKg[e] = T1[(e >> 5) * LDP + (e & 31)];
  __syncthreads();

  mm_valu(T1, Ml, Kl, tid);  // MK = M@K
  __syncthreads();
  for (int e = tid; e < NS * NS; e += 128)
    MKg[e] = T1[(e >> 5) * LDP + (e & 31)];
  __syncthreads();

  mm_valu(T1, T0, Kl, tid);  // CMK = CM@K
  __syncthreads();
  for (int e = tid; e < NS * NS; e += 128)
    CMKg[e] = T1[(e >> 5) * LDP + (e & 31)];
  __syncthreads();

  mm_valu(T1, Ml, Ml, tid);  // M2 = M@M
  __syncthreads();
  for (int e = tid; e < NS * NS; e += 128) {
    const int i = e >> 5, j = e & 31;
    M2g[e] = T1[i * LDP + j];
    R[i * LDP + j] = (i == j) ? 1.f : 0.f;  // R = I
    B[i * LDP + j] = Ml[i * LDP + j];       // B = M
  }
  __syncthreads();

  int e = L;  // ML = M^L (binary exponentiation)
  while (e > 0) {
    if (e & 1) {
      mm_valu(T0, R, B, tid);
      __syncthreads();
      for (int q = tid; q < NS * NS; q += 128) {
        const int i = q >> 5, j = q & 31;
        R[i * LDP + j] = T0[i * LDP + j];
      }
      __syncthreads();
    }
    e >>= 1;
    if (e) {
      mm_valu(T1, B, B, tid);
      __syncthreads();
      for (int q = tid; q < NS * NS; q += 128) {
        const int i = q >> 5, j = q & 31;
        B[i * LDP + j] = T1[i * LDP + j];
      }
      __syncthreads();
    }
  }
  for (int q = tid; q < NS * NS; q += 128) {
    const int i = q >> 5, j = q & 31;
    MLg[q] = R[i * LDP + j];
  }
}

// ============================ Kernel A =====================================
// Per-chunk Horner: c_b = sum_{j in chunk} M^{len-1-j} K o_prev(j)
__global__ __launch_bounds__(32, 1) void kf_horner_kernel(
    const float* __restrict__ obs, const float* __restrict__ K,
    const float* __restrict__ Mg, float* __restrict__ cvec, int T, int L) {
  const int b = blockIdx.x, lane = threadIdx.x;
  const int i0 = b * L;
  const int i1 = (i0 + L < T) ? (i0 + L) : T;
  float Mrow[NS], Krow[NS];
#pragma unroll
  for (int j = 0; j < NS; ++j) {
    Mrow[j] = Mg[lane * NS + j];
    Krow[j] = K[lane * NS + j];
  }
  float c = 0.f;
#pragma unroll 1
  for (int i = i0; i < i1; ++i) {
    const int ip = (i == 0) ? (T - 1) : (i - 1);
    const float op = obs[ip * NS + lane];
    float cn = 0.f;
#pragma unroll
    for (int j = 0; j < NS; ++j)
      cn += Mrow[j] * bcastf(c, j) + Krow[j] * bcastf(op, j);
    c = cn;
  }
  cvec[b * NS + lane] = c;
}

// ============================ Kernel B2 ====================================
// Chunk-start states: x_start (wave0 readlane matvecs), P_start (WMMA chain)
__global__ __launch_bounds__(128, 1) void kf_scanstart_kernel(
    const float* __restrict__ x0, const float* __restrict__ P0,
    const float* __restrict__ MLg, const float* __restrict__ cvec,
    float* __restrict__ xstart, float* __restrict__ Pstart, int NB) {
  __shared__ float Pp[2][NS * LDP];
  const int tid = threadIdx.x;
  const int lane = tid & 31;
  const int w = tid >> 5;
  const int tm = w >> 1, tn = w & 1;
  const int l16 = lane & 15;
  const int lhalf = lane >> 4;

  v2f MLfrag[8];
#pragma unroll
  for (int kc = 0; kc < 8; ++kc) {
    const int r = tm * TILE + l16;
    const int k0 = kc * 4 + 2 * lhalf;
    MLfrag[kc].x = MLg[r * NS + k0];
    MLfrag[kc].y = MLg[r * NS + k0 + 1];
  }
  for (int e = tid; e < NS * NS; e += 128) {
    const int i = e >> 5, j = e & 31;
    Pp[0][i * LDP + j] = P0[e];
  }
  // x chain (wave 0 only)
  if (tid < 32) {
    float MLrow[NS];
#pragma unroll
    for (int j = 0; j < NS; ++j) MLrow[j] = MLg[lane * NS + j];
    float x = x0[lane];
#pragma unroll 1
    for (int b = 0; b < NB; ++b) {
      xstart[b * NS + lane] = x;
      float xn = cvec[b * NS + lane];
#pragma unroll
      for (int j = 0; j < NS; ++j) xn += MLrow[j] * bcastf(x, j);
      x = xn;
    }
  }
  __syncthreads();
  // P chain via WMMA: P_start[b+1] = ML @ P_start[b]
  int p = 0;
#pragma unroll 1
  for (int b = 0; b < NB; ++b) {
    for (int e = tid; e < NS * NS; e += 128) {
      const int i = e >> 5, j = e & 31;
      Pstart[b * 1024 + e] = Pp[p][i * LDP + j];
    }
    v8f np = {0.f, 0.f, 0.f, 0.f, 0.f, 0.f, 0.f, 0.f};
#pragma unroll
    for (int kc = 0; kc < 8; ++kc)
      np = wmma4(MLfrag[kc], loadB(Pp[p], kc, tn, l16, lhalf), np);
    storeD(Pp[p ^ 1], np, tm, tn, l16, lhalf);
    __syncthreads();
    p ^= 1;
  }
}

// ============================ Kernel C =====================================
// Heavy chunk pass: TWO scan steps per barrier, partial yvar per block.
__global__ __launch_bounds__(128, 1) void kf_chunk_kernel(
    const float* __restrict__ obs, const float* __restrict__ K,
    const float* __restrict__ CMg, const float* __restrict__ M2g,
    const float* __restrict__ CM2g, const float* __restrict__ CKg,
    const float* __restrict__ MKg, const float* __restrict__ CMKg,
    const float* __restrict__ xstart, const float* __restrict__ Pstart,
    float* __restrict__ partial, int T, int L) {
  __shared__ float Pb[2][NS * LDP];
  __shared__ float CPa[2][NS * LDP];   // CP of step 1
  __shared__ float CPb2[2][NS * LDP];  // CP of step 2

  const int b = blockIdx.x;
  const int i0 = b * L;
  const int i1 = (i0 + L < T) ? (i0 + L) : T;

  const int tid = threadIdx.x;
  const int lane = tid & 31;
  const int w = tid >> 5;
  const int tm = w >> 1, tn = w & 1;
  const int l16 = lane & 15;
  const int lhalf = lane >> 4;

  for (int e = tid; e < NS * NS; e += 128) {
    const int i = e >> 5, j = e & 31;
    Pb[0][i * LDP + j] = Pstart[b * 1024 + e];
  }

  v2f M2frag[8], CMfrag[8], CM2frag[8];
#pragma unroll
  for (int kc = 0; kc < 8; ++kc) {
    const int r = tm * TILE + l16;
    const int k0 = kc * 4 + 2 * lhalf;
    M2frag[kc].x  = M2g[r * NS + k0];
    M2frag[kc].y  = M2g[r * NS + k0 + 1];
    CMfrag[kc].x  = CMg[r * NS + k0];
    CMfrag[kc].y  = CMg[r * NS + k0 + 1];
    CM2frag[kc].x = CM2g[r * NS + k0];
    CM2frag[kc].y = CM2g[r * NS + k0 + 1];
  }
  // composite-map rows for the x/eps path (batch-start state only)
  float CMr[NS], CKr[NS], CM2r[NS], CMKr[NS], M2r[NS], MKr[NS], Kr[NS];
#pragma unroll
  for (int j = 0; j < NS; ++j) {
    CMr[j]  = CMg[lane * NS + j];
    CKr[j]  = CKg[lane * NS + j];
    CM2r[j] = CM2g[lane * NS + j];
    CMKr[j] = CMKg[lane * NS + j];
    M2r[j]  = M2g[lane * NS + j];
    MKr[j]  = MKg[lane * NS + j];
    Kr[j]   = K[lane * NS + j];
  }
  float xreg = xstart[b * NS + lane];
  v8f yv = {0.f, 0.f, 0.f, 0.f, 0.f, 0.f, 0.f, 0.f};
  __syncthreads();

  int p = 0;
#pragma unroll 1
  for (int i = i0; i < i1; i += 2) {
    const bool has2 = (i + 1 < i1);
    const int ip = (i == 0) ? (T - 1) : (i - 1);
    const int i2 = has2 ? (i + 1) : i;
    const float op1 = obs[ip * NS + lane];   // o_prev for step 1
    const float oc1 = obs[i * NS + lane];    // o_cur step 1 (= o_prev step 2)
    const float oc2 = obs[i2 * NS + lane];   // o_cur step 2
    if (i + 16 < i1) __builtin_prefetch(obs + (i + 16) * NS + lane, 0, 0);

    // One set of B fragments feeds all three products (XDL pipe):
    //   cp1 = CM@P, cp2 = CM2@P, P' = M2@P
    v8f np  = {0.f, 0.f, 0.f, 0.f, 0.f, 0.f, 0.f, 0.f};
    v8f cp1 = {0.f, 0.f, 0.f, 0.f, 0.f, 0.f, 0.f, 0.f};
    v8f cp2 = {0.f, 0.f, 0.f, 0.f, 0.f, 0.f, 0.f, 0.f};
#pragma unroll
    for (int kc = 0; kc < 8; ++kc) {
      const v2f bb = loadB(Pb[p], kc, tn, l16, lhalf);
      cp1 = wmma4(CMfrag[kc], bb, cp1);
      cp2 = wmma4(CM2frag[kc], bb, cp2);
      np  = wmma4(M2frag[kc], bb, np);
    }

    // x/eps via composite maps: one broadcast round, 6 independent FMA chains
    float e1a = oc1, e1b = 0.f;
    float e2a = oc2, e2b = 0.f;
    float xa = 0.f, xb = 0.f;
#pragma unroll
    for (int j = 0; j < NS; j += 2) {
      const float xb0 = bcastf(xreg, j), xb1 = bcastf(xreg, j + 1);
      const float pb0 = bcastf(op1, j),  pb1 = bcastf(op1, j + 1);
      const float cb0 = bcastf(oc1, j),  cb1 = bcastf(oc1, j + 1);
      e1a -= CMr[j] * xb0 + CKr[j] * pb0;
      e1b -= CMr[j + 1] * xb1 + CKr[j + 1] * pb1;
      e2a -= CM2r[j] * xb0 + CMKr[j] * pb0 + CKr[j] * cb0;
      e2b -= CM2r[j + 1] * xb1 + CMKr[j + 1] * pb1 + CKr[j + 1] * cb1;
      xa += M2r[j] * xb0 + MKr[j] * pb0 + Kr[j] * cb0;
      xb += M2r[j + 1] * xb1 + MKr[j + 1] * pb1 + Kr[j + 1] * cb1;
    }
    const float e1 = e1a + e1b;
    const float e2 = e2a + e2b;
    if (has2) xreg = xa + xb;  // x after last batch is dead, skip on tail

    storeD(Pb[p ^ 1], np, tm, tn, l16, lhalf);
    storeD(CPa[p], cp1, tm, tn, l16, lhalf);
    storeD(CPb2[p], cp2, tm, tn, l16, lhalf);
    __syncthreads();   // single barrier per TWO scan steps

    // yvar += CP1@CP1^T + e1 e1^T  (step 1)
#pragma unroll
    for (int kc = 0; kc < 8; ++kc) {
      const int kk = kc * 4 + 2 * lhalf;
      v2f a, bb;
      a.x  = CPa[p][(tm * TILE + l16) * LDP + kk];
      a.y  = CPa[p][(tm * TILE + l16) * LDP + kk + 1];
      bb.x = CPa[p][(tn * TILE + l16) * LDP + kk];
      bb.y = CPa[p][(tn * TILE + l16) * LDP + kk + 1];
      yv = wmma4(a, bb, yv);
    }
    {
      const float ec = __shfl(e1, tn * TILE + l16, 32);
#pragma unroll
      for (int j = 0; j < 8; ++j)
        yv[j] += __shfl(e1, tm * TILE + j + 8 * lhalf, 32) * ec;
    }
    if (has2) {  // step 2 (uniform guard; skipped only on an odd tail)
#pragma unroll
      for (int kc = 0; kc < 8; ++kc) {
        const int kk = kc * 4 + 2 * lhalf;
        v2f a, bb;
        a.x  = CPb2[p][(tm * TILE + l16) * LDP + kk];
        a.y  = CPb2[p][(tm * TILE + l16) * LDP + kk + 1];
        bb.x = CPb2[p][(tn * TILE + l16) * LDP + kk];
        bb.y = CPb2[p][(tn * TILE + l16) * LDP + kk + 1];
        yv = wmma4(a, bb, yv);
      }
      const float ec = __shfl(e2, tn * TILE + l16, 32);
#pragma unroll
      for (int j = 0; j < 8; ++j)
        yv[j] += __shfl(e2, tm * TILE + j + 8 * lhalf, 32) * ec;
    }
    p ^= 1;
  }

#pragma unroll
  for (int j = 0; j < 8; ++j)
    partial[b * 1024 + (tm * TILE + j + 8 * lhalf) * NS + tn * TILE + l16] =
        yv[j];
}

// ============================ Kernel D =====================================
// Fixed-order reduction + initial term, /T, Cholesky, 2*sum(log diag).
__global__ __launch_bounds__(128, 1) void kf_final_kernel(
    const float* __restrict__ obs, const float* __restrict__ C,
    const float* __restrict__ P0, const float* __restrict__ partial,
    float* __restrict__ out, int T, int NB) {
  __shared__ float Y[NS * LDP], CP0[NS * LDP], Pl[NS * LDP], Cl[NS * LDP];
  const int tid = threadIdx.x;
  for (int e = tid; e < NS * NS; e += 128) {
    const int i = e >> 5, j = e & 31;
    Pl[i * LDP + j] = P0[e];
    Cl[i * LDP + j] = C[e];
  }
  __syncthreads();
  mm_valu(CP0, Cl, Pl, tid);   // CP0 = C @ Psqrt0
  __syncthreads();
  const float invT = 1.0f / (float)T;
  for (int e = tid; e < NS * NS; e += 128) {
    const int i = e >> 5, j = e & 31;
    float s = 0.f;
    for (int b = 0; b < NB; ++b) s += partial[b * 1024 + e];
    s += obs[i] * obs[j];                       // outer(obs[0], obs[0])
    float t = 0.f;
#pragma unroll 8
    for (int k = 0; k < NS; ++k) t += CP0[i * LDP + k] * CP0[j * LDP + k];
    Y[i * LDP + j] = (s + t) * invT;
  }
  __syncthreads();
  for (int kstep = 0; kstep < NS; ++kstep) {
    if (tid == 0) Y[kstep * LDP + kstep] = sqrtf(Y[kstep * LDP + kstep]);
    __syncthreads();
    if (tid > kstep && tid < NS)
      Y[tid * LDP + kstep] /= Y[kstep * LDP + kstep];
    __syncthreads();
    for (int e = tid; e < NS * NS; e += 128) {
      const int ii = e >> 5, jj = e & 31;
      if (ii > kstep && jj > kstep && jj <= ii)
        Y[ii * LDP + jj] -= Y[ii * LDP + kstep] * Y[jj * LDP + kstep];
    }
    __syncthreads();
  }
  if (tid == 0) {
    float s = 0.f;
#pragma unroll
    for (int k = 0; k < NS; ++k) s += __logf(Y[k * LDP + k]);
    out[0] = 2.0f * s;
  }
}

// =================== Serial fallback (single block) ========================
__global__ __launch_bounds__(128, 1) void kf_scan_kernel(
    const float* __restrict__ obs, const float* __restrict__ A,
    const float* __restrict__ C, const float* __restrict__ K,
    const float* __restrict__ x0, const float* __restrict__ P0,
    float* __restrict__ out, int T) {
  __shared__ float Mlds[NS * LDP];
  __shared__ float Pb[2][NS * LDP];
  __shared__ float CPb[NS * LDP];

  const int tid = threadIdx.x;
  const int lane = tid & 31;
  const int w = tid >> 5;
  const int tm = w >> 1, tn = w & 1;
  const int l16 = lane & 15;
  const int lhalf = lane >> 4;

  for (int e = tid; e < NS * NS; e += 128) {
    const int i = e >> 5, j = e & 31;
    float acc = A[i * NS + j];
#pragma unroll 8
    for (int k = 0; k < NS; ++k) acc -= K[i * NS + k] * C[k * NS + j];
    Mlds[i * LDP + j] = acc;
    Pb[0][i * LDP + j] = P0[i * NS + j];
  }
  __syncthreads();

  v2f Mfrag[8], Cfrag[8];
#pragma unroll
  for (int kc = 0; kc < 8; ++kc) {
    const int r = tm * TILE + l16;
    const int k0 = kc * 4 + 2 * lhalf;
    Mfrag[kc].x = Mlds[r * LDP + k0];
    Mfrag[kc].y = Mlds[r * LDP + k0 + 1];
    Cfrag[kc].x = C[r * NS + k0];
    Cfrag[kc].y = C[r * NS + k0 + 1];
  }
  float Mrow[NS], Krow[NS], Crow[NS];
#pragma unroll
  for (int j = 0; j < NS; ++j) {
    Mrow[j] = Mlds[lane * LDP + j];
    Krow[j] = K[lane * NS + j];
    Crow[j] = C[lane * NS + j];
  }
  float xreg = x0[lane];
  v8f yv = {0.f, 0.f, 0.f, 0.f, 0.f, 0.f, 0.f, 0.f};

  auto accumYvar = [&](float epsv) {
#pragma unroll
    for (int kc = 0; kc < 8; ++kc) {
      const int kk = kc * 4 + 2 * lhalf;
      v2f a, bb;
      a.x = CPb[(tm * TILE + l16) * LDP + kk];
      a.y = CPb[(tm * TILE + l16) * LDP + kk + 1];
      bb.x = CPb[(tn * TILE + l16) * LDP + kk];
      bb.y = CPb[(tn * TILE + l16) * LDP + kk + 1];
      yv = wmma4(a, bb, yv);
    }
    const float ec = __shfl(epsv, tn * TILE + l16, 32);
#pragma unroll
    for (int j = 0; j < 8; ++j)
      yv[j] += __shfl(epsv, tm * TILE + j + 8 * lhalf, 32) * ec;
  };

  {
    v8f cp = {0.f, 0.f, 0.f, 0.f, 0.f, 0.f, 0.f, 0.f};
#pragma unroll
    for (int kc = 0; kc < 8; ++kc)
      cp = wmma4(Cfrag[kc], loadB(Pb[0], kc, tn, l16, lhalf), cp);
    storeD(CPb, cp, tm, tn, l16, lhalf);
  }
  const float eps0 = obs[lane];
  __syncthreads();
  accumYvar(eps0);

  int cur = 0;
#pragma unroll 1
  for (int i = 0; i < T; ++i) {
    const int ip = (i == 0) ? (T - 1) : (i - 1);
    const float op = obs[ip * NS + lane];
    const float oc = obs[i * NS + lane];
    if (i + 16 < T) __builtin_prefetch(obs + (i + 16) * NS + lane, 0, 0);

    v8f np = {0.f, 0.f, 0.f, 0.f, 0.f, 0.f, 0.f, 0.f};
#pragma unroll
    for (int kc = 0; kc < 8; ++kc)
      np = wmma4(Mfrag[kc], loadB(Pb[cur], kc, tn, l16, lhalf), np);

    float xn = 0.f;
#pragma unroll
    for (int j = 0; j < NS; ++j)
      xn += Mrow[j] * bcastf(xreg, j) + Krow[j] * bcastf(op, j);
    float ea = oc;
#pragma unroll
    for (int j = 0; j < NS; ++j) ea -= Crow[j] * bcastf(xn, j);
    xreg = xn;

    storeD(Pb[cur ^ 1], np, tm, tn, l16, lhalf);
    __syncthreads();
    cur ^= 1;

    v8f cp = {0.f, 0.f, 0.f, 0.f, 0.f, 0.f, 0.f, 0.f};
#pragma unroll
    for (int kc = 0; kc < 8; ++kc)
      cp = wmma4(Cfrag[kc], loadB(Pb[cur], kc, tn, l16, lhalf), cp);
    storeD(CPb, cp, tm, tn, l16, lhalf);
    __syncthreads();

    accumYvar(ea);
  }

  __syncthreads();
  const float invT = 1.0f / (float)T;
#pragma unroll
  for (int j = 0; j < 8; ++j)
    CPb[(tm * TILE + j + 8 * lhalf) * LDP + tn * TILE + l16] = yv[j] * invT;
  __syncthreads();

  for (int kstep = 0; kstep < NS; ++kstep) {
    if (tid == 0) CPb[kstep * LDP + kstep] = sqrtf(CPb[kstep * LDP + kstep]);
    __syncthreads();
    if (tid > kstep && tid < NS)
      CPb[tid * LDP + kstep] /= CPb[kstep * LDP + kstep];
    __syncthreads();
    for (int e = tid; e < NS * NS; e += 128) {
      const int ii = e >> 5, jj = e & 31;
      if (ii > kstep && jj > kstep && jj <= ii)
        CPb[ii * LDP + jj] -= CPb[ii * LDP + kstep] * CPb[jj * LDP + kstep];
    }
    __syncthreads();
  }
  if (tid == 0) {
    float s = 0.f;
#pragma unroll
    for (int k = 0; k < NS; ++k) s += __logf(CPb[k * LDP + k]);
    out[0] = 2.0f * s;
  }
}

// =========================== host launcher =================================
extern "C" void kernel_launch(void* const* d_in, const int* in_sizes, int n_in,
                              void* d_out, int out_size, void* d_ws,
                              size_t ws_size, hipStream_t stream) {
  const float* obs = (const float*)d_in[0];  // [T, 32]
  const float* A   = (const float*)d_in[1];  // [32, 32]
  const float* C   = (const float*)d_in[2];  // [32, 32]
  const float* K   = (const float*)d_in[3];  // [32, 32]
  const float* x0  = (const float*)d_in[4];  // [32]
  const float* P0  = (const float*)d_in[5];  // [32, 32]
  float* out = (float*)d_out;
  const int T = in_sizes[0] / NS;

  int NB = 250;
  int L = (T + NB - 1) / NB;
  if (L < 1) L = 1;
  NB = (T + L - 1) / L;  // drop empty chunks

  const size_t need =
      (size_t)(8 * 1024 + (size_t)NB * (32 + 32 + 1024 + 1024)) * sizeof(float);

  if (T < 2 * NB || ws_size < need) {
    // deterministic fallback: single-block serial scan
    kf_scan_kernel<<<1, 128, 0, stream>>>(obs, A, C, K, x0, P0, out, T);
    return;
  }

  float* Mg      = (float*)d_ws;
  float* CMg     = Mg + 1024;
  float* M2g     = CMg + 1024;
  float* CM2g    = M2g + 1024;
  float* CKg     = CM2g + 1024;
  float* MKg     = CKg + 1024;
  float* CMKg    = MKg + 1024;
  float* MLg     = CMKg + 1024;
  float* cvec    = MLg + 1024;
  float* xstart  = cvec + (size_t)NB * 32;
  float* Pstart  = xstart + (size_t)NB * 32;
  float* partial = Pstart + (size_t)NB * 1024;

  kf_prep_kernel<<<1, 128, 0, stream>>>(A, C, K, Mg, CMg, M2g, CM2g, CKg, MKg,
                                        CMKg, MLg, L);
  kf_horner_kernel<<<NB, 32, 0, stream>>>(obs, K, Mg, cvec, T, L);
  kf_scanstart_kernel<<<1, 128, 0, stream>>>(x0, P0, MLg, cvec, xstart, Pstart,
                                             NB);
  kf_chunk_kernel<<<NB, 128, 0, stream>>>(obs, K, CMg, M2g, CM2g, CKg, MKg,
                                          CMKg, xstart, Pstart, partial, T, L);
  kf_final_kernel<<<1, 128, 0, stream>>>(obs, C, P0, partial, out, T, NB);
}